// GraphConvLayer_49194555408403
// MI455X (gfx1250) — compile-verified
//
#include <hip/hip_runtime.h>

#define S_   256
#define N_   558
#define B2_  8192
#define F_   128
#define M_   (S_ * N_)   // 142848 total nodes
#define TE_  (S_ * B2_)  // 2097152 total edges

typedef __attribute__((ext_vector_type(2))) float v2f;
typedef __attribute__((ext_vector_type(8))) float v8f;

// ---------------- K0: init deg = 1.0 (self loop), bd_max = 0 ----------------
__global__ __launch_bounds__(256) void k_init(float* __restrict__ deg,
                                              int* __restrict__ bdmax) {
    int i = blockIdx.x * 256 + threadIdx.x;
    if (i < M_) deg[i] = 1.0f;
    if (i == 0) *bdmax = 0;
}

// ---------------- K1: masked global max of bond_dist ----------------
__global__ __launch_bounds__(256) void k_bdmax(const int* __restrict__ ei,
                                               const float* __restrict__ bd,
                                               int* __restrict__ bdmax) {
    int stride = gridDim.x * 256;
    float m = 0.0f;
    for (int i = blockIdx.x * 256 + threadIdx.x; i < TE_; i += stride) {
        int s    = i >> 13;             // B2 = 8192 = 2^13
        int base = i + (s << 13);       // ei[s][0][e]
        int src  = ei[base];
        int dst  = ei[base + B2_];      // ei[s][1][e]
        if (src != dst) m = fmaxf(m, bd[i]);
    }
    // wave32 butterfly reduce
    for (int off = 16; off > 0; off >>= 1)
        m = fmaxf(m, __shfl_xor(m, off, 32));
    // values are all >= 0, so int-pattern max == float max
    if ((threadIdx.x & 31) == 0) atomicMax(bdmax, __float_as_int(m));
}

// ---------------- K2: degree accumulation ----------------
__global__ __launch_bounds__(256) void k_deg(const int* __restrict__ ei,
                                             const float* __restrict__ bd,
                                             const int* __restrict__ bdmax,
                                             float* __restrict__ deg) {
    int i = blockIdx.x * 256 + threadIdx.x;
    if (i >= TE_) return;
    int s    = i >> 13;
    int base = i + (s << 13);
    int src  = ei[base];
    int dst  = ei[base + B2_];
    if (src == dst) return;
    float bm = __int_as_float(*bdmax);
    unsafeAtomicAdd(&deg[s * N_ + dst], bd[i] / bm);
}

// ---------------- K3: dinv = rsqrt(deg); agg = (1/deg) * x (self-loop term) ----------------
__global__ __launch_bounds__(256) void k_dinv_agg(const float* __restrict__ x,
                                                  const float* __restrict__ deg,
                                                  float* __restrict__ dinv,
                                                  float* __restrict__ agg) {
    int t = blockIdx.x * 256 + threadIdx.x;   // one wave (32 lanes x float4) per node row
    int m = t >> 5;
    if (m >= M_) return;
    int f4   = (t & 31) * 4;
    float d  = deg[m];
    float di = rsqrtf(d);                 // deg >= 1 always
    if ((t & 31) == 0) dinv[m] = di;
    float sl = di * di;                   // self-loop norm = 1/deg
    const float4 xv = *(const float4*)(x + (size_t)m * F_ + f4);
    float4 o;
    o.x = xv.x * sl; o.y = xv.y * sl; o.z = xv.z * sl; o.w = xv.w * sl;
    *(float4*)(agg + (size_t)m * F_ + f4) = o;
}

// ---------------- K4: edge scatter, one wave per edge, atomics land in L2 ----------------
__global__ __launch_bounds__(256) void k_scatter(const int* __restrict__ ei,
                                                 const float* __restrict__ bd,
                                                 const float* __restrict__ x,
                                                 const float* __restrict__ dinv,
                                                 const int* __restrict__ bdmax,
                                                 float* __restrict__ agg) {
    int lane = threadIdx.x & 31;
    int e    = blockIdx.x * 8 + (threadIdx.x >> 5);
    if (e >= TE_) return;
    int s    = e >> 13;
    int base = e + (s << 13);
    int src  = ei[base];
    int dst  = ei[base + B2_];
    if (src == dst) return;
    int row = s * N_ + src;
    int col = s * N_ + dst;
    float bm   = __int_as_float(*bdmax);
    float norm = dinv[row] * (bd[e] / bm) * dinv[col];
    const float4 xv = *(const float4*)(x + (size_t)row * F_ + lane * 4);
    float* dp = agg + (size_t)col * F_ + lane * 4;
    unsafeAtomicAdd(dp + 0, norm * xv.x);
    unsafeAtomicAdd(dp + 1, norm * xv.y);
    unsafeAtomicAdd(dp + 2, norm * xv.z);
    unsafeAtomicAdd(dp + 3, norm * xv.w);
}

// ---------------- K5: out = agg @ W^T + bias via V_WMMA_F32_16X16X4_F32 ----------------
// A (16x4 f32): lanes 0-15 hold M=lane, {K=k,K=k+1}; lanes 16-31 hold M=lane-16, {K=k+2,K=k+3}
// B (4x16 f32): lanes 0-15 hold N=lane,  {K=k,K=k+1}; lanes 16-31 hold N=lane-16, {K=k+2,K=k+3}
// D (16x16 f32): VGPR v -> M = m0 + v + (lane>=16 ? 8 : 0), N = lane & 15
__global__ __launch_bounds__(256) void k_gemm(const float* __restrict__ agg,
                                              const float* __restrict__ W,
                                              const float* __restrict__ bias,
                                              float* __restrict__ out) {
    int lane = threadIdx.x & 31;
    int wave = threadIdx.x >> 5;
    int m0   = blockIdx.x * 16;
    int o0   = wave * 16;
    int l16  = lane & 15;
    int koff = (lane >> 4) * 2;

    const float* arow = agg + (size_t)(m0 + l16) * F_ + koff;
    const float* brow = W   + (size_t)(o0 + l16) * F_ + koff;   // B[k][o] = W[o][k]

    v8f acc = {};
#pragma unroll
    for (int k = 0; k < F_; k += 4) {
        v2f a = *(const v2f*)(arow + k);
        v2f b = *(const v2f*)(brow + k);
        acc = __builtin_amdgcn_wmma_f32_16x16x4_f32(
            /*neg_a=*/false, a, /*neg_b=*/false, b,
            /*c_mod=*/(short)0, acc, /*reuse_a=*/false, /*reuse_b=*/false);
    }

    float bb    = bias[o0 + l16];
    int   half8 = (lane >> 4) * 8;
#pragma unroll
    for (int v = 0; v < 8; ++v) {
        out[(size_t)(m0 + v + half8) * F_ + o0 + l16] = acc[v] + bb;
    }
}

extern "C" void kernel_launch(void* const* d_in, const int* in_sizes, int n_in,
                              void* d_out, int out_size, void* d_ws, size_t ws_size,
                              hipStream_t stream) {
    const float* x    = (const float*)d_in[0];   // (S, N, 128) f32
    const int*   ei   = (const int*)  d_in[1];   // (S, 2, B2) i32
    const float* bd   = (const float*)d_in[2];   // (S, B2)    f32
    const float* W    = (const float*)d_in[3];   // (128, 128) f32
    const float* bias = (const float*)d_in[4];   // (128,)     f32
    float*       out  = (float*)d_out;           // (S, N, 128) f32

    // workspace layout: agg (M*128 f32) | deg (M f32) | dinv (M f32) | bdmax (1 i32)
    float* agg   = (float*)d_ws;
    float* deg   = agg + (size_t)M_ * F_;
    float* dinv  = deg + M_;
    int*   bdmax = (int*)(dinv + M_);

    k_init    <<<(M_ + 255) / 256, 256, 0, stream>>>(deg, bdmax);
    k_bdmax   <<<2048,            256, 0, stream>>>(ei, bd, bdmax);
    k_deg     <<<TE_ / 256,       256, 0, stream>>>(ei, bd, bdmax, deg);
    k_dinv_agg<<<(M_ * 32) / 256, 256, 0, stream>>>(x, deg, dinv, agg);
    k_scatter <<<TE_ / 8,         256, 0, stream>>>(ei, bd, x, dinv, bdmax, agg);
    k_gemm    <<<M_ / 16,         256, 0, stream>>>(agg, W, bias, out);
}